// EmptyImageDetector_27049704030576
// MI455X (gfx1250) — compile-verified
//
#include <hip/hip_runtime.h>
#include <stdint.h>

// ---------------------------------------------------------------------------
// EmptyImageDetector for MI455X (gfx1250, wave32)
//   Stage A: per-(batch,channel) sum / sum-of-squares via V_WMMA_F32_16X16X4_F32
//            used as an fp32 reduction tree (B = ones -> D rows = row sums).
//   Stage B: exact-in-practice unique pixel count via lock-free 64-bit
//            fingerprint hash set (single-word atomicCAS, no intra-wave spin).
//   Stage C: finalize -> [unique(32) | color_var(32) | brightness(32)] floats.
// ---------------------------------------------------------------------------

typedef float v2f __attribute__((ext_vector_type(2)));
typedef float v8f __attribute__((ext_vector_type(8)));

#define STATS_THREADS 256
#define STATS_WAVES   8

#define TABLE_BITS 19
#define TABLE_SIZE (1u << TABLE_BITS)          // 524288 slots
#define TABLE_MASK (TABLE_SIZE - 1u)

// ---------------------------------------------------------------------------
// Stage A: one block per (batch,channel) plane. 8 waves, each wave reduces a
// contiguous N/8 slice. Inner loop: 64 coalesced floats per wave -> 2 WMMAs.
// ---------------------------------------------------------------------------
__global__ void __launch_bounds__(STATS_THREADS)
stats_kernel(const float* __restrict__ in, float* __restrict__ stats, int N) {
    const int p = blockIdx.x;                         // plane id 0..B*C-1
    const float* base = in + (size_t)p * (size_t)N;

    const int wave = threadIdx.x >> 5;
    const int lane = threadIdx.x & 31;

    const int perWave     = N / STATS_WAVES;          // N is 262144 -> 32768
    const int mainPerWave = (perWave / 64) * 64;      // full 64-float chunks
    const float* wbase = base + (size_t)wave * (size_t)perWave;

    v8f c_sum = {0.f, 0.f, 0.f, 0.f, 0.f, 0.f, 0.f, 0.f};
    v8f c_sq  = {0.f, 0.f, 0.f, 0.f, 0.f, 0.f, 0.f, 0.f};
    v2f ones;  ones.x = 1.0f; ones.y = 1.0f;

    for (int it = 0; it < mainPerWave; it += 64) {
        const float* pp = wbase + it + lane * 2;
        v2f a;  a.x = pp[0];        a.y = pp[1];
        v2f a2; a2.x = a.x * a.x;   a2.y = a.y * a.y;
        // D = A(16x4,f32) * ones(4x16) + C  -> fp32 row-sum accumulate
        c_sum = __builtin_amdgcn_wmma_f32_16x16x4_f32(
            false, a,  false, ones, (short)0, c_sum, false, false);
        c_sq  = __builtin_amdgcn_wmma_f32_16x16x4_f32(
            false, a2, false, ones, (short)0, c_sq,  false, false);
    }

    // Collapse C tile: lanes 0-15 hold rows 0-7, lanes 16-31 hold rows 8-15.
    float s = c_sum[0] + c_sum[1] + c_sum[2] + c_sum[3] +
              c_sum[4] + c_sum[5] + c_sum[6] + c_sum[7];
    float q = c_sq[0] + c_sq[1] + c_sq[2] + c_sq[3] +
              c_sq[4] + c_sq[5] + c_sq[6] + c_sq[7];
    s += __shfl_xor(s, 16, 32);
    q += __shfl_xor(q, 16, 32);

    // Scalar tail (never taken for N = 262144, kept for generality).
    if (mainPerWave < perWave) {
        float ts = 0.f, tq = 0.f;
        for (int i = mainPerWave + lane; i < perWave; i += 32) {
            float v = wbase[i];
            ts += v; tq += v * v;
        }
        for (int off = 16; off >= 1; off >>= 1) {
            ts += __shfl_xor(ts, off, 32);
            tq += __shfl_xor(tq, off, 32);
        }
        s += ts; q += tq;
    }

    __shared__ float sh_s[STATS_WAVES];
    __shared__ float sh_q[STATS_WAVES];
    if (lane == 0) { sh_s[wave] = s; sh_q[wave] = q; }
    __syncthreads();
    if (threadIdx.x == 0) {
        float S = 0.f, Q = 0.f;
        for (int w = 0; w < STATS_WAVES; ++w) { S += sh_s[w]; Q += sh_q[w]; }
        stats[2 * p + 0] = S;
        stats[2 * p + 1] = Q;
    }
}

// ---------------------------------------------------------------------------
// Stage B helpers
// ---------------------------------------------------------------------------
__device__ __forceinline__ unsigned long long splitmix64(unsigned long long x) {
    x += 0x9E3779B97F4A7C15ULL;
    x = (x ^ (x >> 30)) * 0xBF58476D1CE4E5B9ULL;
    x = (x ^ (x >> 27)) * 0x94D049BB133111EBULL;
    return x ^ (x >> 31);
}

__global__ void clear_u64(unsigned long long* __restrict__ t, size_t n) {
    size_t i      = (size_t)blockIdx.x * blockDim.x + threadIdx.x;
    size_t stride = (size_t)gridDim.x * blockDim.x;
    for (; i < n; i += stride) t[i] = 0ULL;
}

__global__ void clear_u32(unsigned int* __restrict__ c, int n) {
    int i = blockIdx.x * blockDim.x + threadIdx.x;
    if (i < n) c[i] = 0u;
}

__global__ void unique_insert(const unsigned int* __restrict__ rp,
                              const unsigned int* __restrict__ gp,
                              const unsigned int* __restrict__ bp,
                              unsigned long long* __restrict__ table,
                              unsigned int* __restrict__ count, int N) {
    int i = blockIdx.x * blockDim.x + threadIdx.x;
    if (i >= N) return;

    unsigned long long x = ((unsigned long long)rp[i] << 32) | (unsigned long long)gp[i];
    x = splitmix64(x);
    x = splitmix64(x ^ (((unsigned long long)bp[i] << 1) | 1ULL));
    if (x == 0ULL) x = 1ULL;                        // 0 reserved for empty

    unsigned int slot = (unsigned int)x & TABLE_MASK;
    for (;;) {
        unsigned long long prev = atomicCAS(table + slot, 0ULL, x);
        if (prev == 0ULL) { atomicAdd(count, 1u); return; }   // newly inserted
        if (prev == x)    return;                             // duplicate
        slot = (slot + 1u) & TABLE_MASK;                      // linear probe
    }
}

// ---------------------------------------------------------------------------
// Stage C: outputs (all float): [unique 0..31 | colvar 32..63 | bright 64..95]
// ---------------------------------------------------------------------------
__global__ void finalize_kernel(const float* __restrict__ stats,
                                const unsigned int* __restrict__ counts,
                                float* __restrict__ out, int N, int B) {
    int b = blockIdx.x * blockDim.x + threadIdx.x;
    if (b >= B) return;
    const float Nf = (float)N;
    float var = 0.f, tot = 0.f;
    for (int c = 0; c < 3; ++c) {
        float s = stats[2 * (b * 3 + c) + 0];
        float q = stats[2 * (b * 3 + c) + 1];
        var += (q - (s * s) / Nf) / (Nf - 1.0f);    // ddof=1
        tot += s;
    }
    out[b]          = (float)counts[b];             // unique color count
    out[B + b]      = var * (1.0f / 3.0f);          // mean channel variance
    out[2 * B + b]  = tot / (3.0f * Nf);            // brightness
}

// ---------------------------------------------------------------------------
// Launch
// ---------------------------------------------------------------------------
extern "C" void kernel_launch(void* const* d_in, const int* in_sizes, int n_in,
                              void* d_out, int out_size, void* d_ws, size_t ws_size,
                              hipStream_t stream) {
    const float* in = (const float*)d_in[0];
    const int B = 32, C = 3;
    const long long total = (long long)in_sizes[0];
    const int N = (int)(total / (long long)(B * C));     // 262144

    // Workspace layout:
    //   [0, 768)        : stats, 96 planes x {sum, sumsq} (float)
    //   [1024, 1152)    : per-batch unique counters (u32)
    //   [4096, ...)     : hash tables, TABLE_SIZE u64 each (4 MB / batch)
    char* ws = (char*)d_ws;
    float*              stats  = (float*)ws;
    unsigned int*       counts = (unsigned int*)(ws + 1024);
    unsigned long long* tables = (unsigned long long*)(ws + 4096);

    const size_t TBYTES = (size_t)TABLE_SIZE * 8ULL;
    int nTables = 1;
    if (ws_size > 4096 + TBYTES) {
        size_t t = (ws_size - 4096) / TBYTES;
        nTables = (t > (size_t)B) ? B : (int)t;
        if (nTables < 1) nTables = 1;
    }

    // Zero counters, compute per-plane sum/sumsq (WMMA reduction).
    clear_u32<<<1, 64, 0, stream>>>(counts, B);
    stats_kernel<<<B * C, STATS_THREADS, 0, stream>>>(in, stats, N);

    // Unique counts: passes of nTables batches, stream-serialized.
    const int insBlocks = (N + 255) / 256;
    for (int b0 = 0; b0 < B; b0 += nTables) {
        int nb = B - b0; if (nb > nTables) nb = nTables;
        clear_u64<<<2048, 256, 0, stream>>>(tables, (size_t)nb * TABLE_SIZE);
        for (int j = 0; j < nb; ++j) {
            int b = b0 + j;
            const unsigned int* rp = (const unsigned int*)(in + ((size_t)b * 3 + 0) * (size_t)N);
            const unsigned int* gp = (const unsigned int*)(in + ((size_t)b * 3 + 1) * (size_t)N);
            const unsigned int* bp = (const unsigned int*)(in + ((size_t)b * 3 + 2) * (size_t)N);
            unique_insert<<<insBlocks, 256, 0, stream>>>(
                rp, gp, bp, tables + (size_t)j * TABLE_SIZE, counts + b, N);
        }
    }

    finalize_kernel<<<1, 64, 0, stream>>>(stats, counts, (float*)d_out, N, B);
}